// Encoder_1666447310976
// MI455X (gfx1250) — compile-verified
//
#include <hip/hip_runtime.h>
#include <hip/hip_bf16.h>
#include <math.h>
#include <stdint.h>

// Problem constants (fixed by the reference harness)
#define B_   16
#define HW_  1024          // 32*32 spatial positions
#define C_   256
#define K_   32
#define ROWS 64            // spatial rows per workgroup tile
#define TILES_PER_BATCH (HW_ / ROWS)   // 16
#define NBLK1 (B_ * TILES_PER_BATCH)   // 256 workgroups
#define BN_EPS 1e-3f

typedef float v2f  __attribute__((ext_vector_type(2)));
typedef float v8f  __attribute__((ext_vector_type(8)));
typedef unsigned int u32x4 __attribute__((ext_vector_type(4)));
typedef unsigned int u32x8 __attribute__((ext_vector_type(8)));

// Workspace layout (in floats)
#define WS_ENC   0
#define WS_WSUM  (B_ * K_ * C_)        // 131072
#define WS_SUM   (WS_WSUM + B_ * K_)   // 131584
#define WS_SQ    (WS_SUM + C_)         // 131840
#define WS_TOTAL (WS_SQ + C_)          // 132096 floats = 528 KB

// ---------------------------------------------------------------------------
// Tensor Data Mover: one-shot 2D tile DMA global -> LDS.
// Builds D# group0/group1 per CDNA5 ISA §8.3/8.4 and issues the 2-group form
// (VADDR2/VADDR3 = NULL => tensors up to 2D). Tracked by TENSORcnt.
//   rows x cols f32 tile, row stride = strideElems elements.
// ---------------------------------------------------------------------------
__device__ __forceinline__ void tdm_load_2d_f32(const void* gsrc, void* ldst,
                                                unsigned rows, unsigned cols,
                                                unsigned strideElems)
{
    unsigned long long ga = (unsigned long long)(uintptr_t)gsrc;
    unsigned ldsOff = (unsigned)(uintptr_t)ldst;   // flat->LDS: low 32 bits

    u32x4 g0;
    g0[0] = 1u;                                    // count=1, user descriptor
    g0[1] = ldsOff;                                // lds_addr (bytes)
    g0[2] = (unsigned)ga;                          // global_addr[31:0]
    g0[3] = (unsigned)(ga >> 32) | (2u << 30);     // global_addr[56:32] | type=2

    u32x8 g1;
    g1[0] = 2u << 16;                              // wg_mask=0, data_size=2 (4B)
    g1[1] = (cols & 0xffffu) << 16;                // tensor_dim0[15:0] @ bits 63:48
    g1[2] = ((cols >> 16) & 0xffffu)               // tensor_dim0[31:16]
          | ((rows & 0xffffu) << 16);              // tensor_dim1[15:0]
    g1[3] = ((rows >> 16) & 0xffffu)               // tensor_dim1[31:16]
          | ((cols & 0xffffu) << 16);              // tile_dim0
    g1[4] = rows & 0xffffu;                        // tile_dim1 (tile_dim2=0)
    g1[5] = strideElems;                           // tensor_dim0_stride[31:0]
    g1[6] = 0u;                                    // stride hi / dim1_stride lo
    g1[7] = 0u;                                    // dim1_stride hi (unused, 2D)

    asm volatile("tensor_load_to_lds %0, %1" :: "s"(g0), "s"(g1) : "memory");
}

// ---------------------------------------------------------------------------
// k0: zero the workspace accumulators (harness does not re-zero between runs)
// ---------------------------------------------------------------------------
__global__ void k0_zero(float* __restrict__ ws, int n) {
    int i = blockIdx.x * blockDim.x + threadIdx.x;
    if (i < n) ws[i] = 0.0f;
}

// ---------------------------------------------------------------------------
// k1: per-tile fused kernel.
//   TDM-stage X tile [ROWS x C] and codewords [K x C] into LDS (one HBM read),
//   GEMM1: dot = X * cw^T via v_wmma_f32_16x16x4_f32,
//   smoothed softmax over K per row,
//   GEMM2: encPart = W^T * X via v_wmma_f32_16x16x4_f32,
//   atomic-accumulate encPart and column sums of W into global workspace.
// ---------------------------------------------------------------------------
__global__ __launch_bounds__(256)
void k1_assign(const float* __restrict__ x,     // [B, HW, C]
               const float* __restrict__ cwg,   // [K, C]
               const float* __restrict__ sfg,   // [K]
               float* __restrict__ enc,         // [B, K, C] accumulator
               float* __restrict__ wsum)        // [B, K] accumulator
{
    extern __shared__ float smem[];
    float* Xt   = smem;                    // ROWS*C      (16384 f)
    float* cw   = Xt   + ROWS * C_;        // K*C         ( 8192 f)
    float* dotb = cw   + K_   * C_;        // ROWS*K      ( 2048 f)
    float* wb   = dotb + ROWS * K_;        // ROWS*K      ( 2048 f)
    float* xn   = wb   + ROWS * K_;        // ROWS*4      (  256 f)
    float* cwn  = xn   + ROWS * 4;         // K
    float* sfl  = cwn  + K_;               // K

    const int tid  = threadIdx.x;
    const int lane = tid & 31;
    const int wave = tid >> 5;                 // 8 waves (wave32)
    const int b    = blockIdx.x / TILES_PER_BATCH;
    const int tile = blockIdx.x % TILES_PER_BATCH;
    const float* gX = x + ((long)b * HW_ + (long)tile * ROWS) * C_;

    // ---- Stage X tile and codewords via Tensor Data Mover (wave 0 only;
    //      TDM ignores EXEC but is issued per wave). ----
    if (wave == 0) {
        tdm_load_2d_f32(gX,  Xt, ROWS, C_, C_);
        tdm_load_2d_f32(cwg, cw, K_,   C_, C_);
        __builtin_amdgcn_s_wait_tensorcnt(0);
    }
    __syncthreads();

    // ---- Partial ||x||^2 per row (4 partials per row) ----
    {
        int r = tid >> 2, p = tid & 3;
        const float* xr = Xt + r * C_ + p * 64;
        float s = 0.0f;
        #pragma unroll 8
        for (int c = 0; c < 64; ++c) { float v = xr[c]; s += v * v; }
        xn[r * 4 + p] = s;
    }
    // ---- ||mu_k||^2 and smoothing factors ----
    if (tid < K_) {
        const float* cr = cw + tid * C_;
        float s = 0.0f;
        #pragma unroll 8
        for (int c = 0; c < C_; ++c) s += cr[c] * cr[c];
        cwn[tid] = s;
        sfl[tid] = sfg[tid];
    }

    // ---- GEMM1: dot[ROWS x K] = Xt[ROWS x C] * cw^T[C x K] ----
    // 4 M-tiles x 2 N-tiles -> one 16x16 tile per wave.
    {
        const int mT = wave >> 1;              // 0..3
        const int nT = wave & 1;               // 0..1
        const int rowA = mT * 16 + (lane & 15);
        const int colB = nT * 16 + (lane & 15);
        const int kHi  = (lane >> 4) * 2;      // lanes 0-15: K{0,1}; 16-31: K{2,3}
        v8f acc = {};
        for (int kk = 0; kk < C_; kk += 4) {
            v2f a, bf;
            a.x  = Xt[rowA * C_ + kk + kHi];
            a.y  = Xt[rowA * C_ + kk + kHi + 1];
            bf.x = cw[colB * C_ + kk + kHi];
            bf.y = cw[colB * C_ + kk + kHi + 1];
            acc = __builtin_amdgcn_wmma_f32_16x16x4_f32(
                false, a, false, bf, (short)0, acc, false, false);
        }
        // C/D layout: VGPR v, lanes 0-15 -> M=v ; lanes 16-31 -> M=v+8
        #pragma unroll
        for (int v = 0; v < 8; ++v) {
            int m = mT * 16 + v + ((lane >> 4) << 3);
            int n = nT * 16 + (lane & 15);
            dotb[m * K_ + n] = acc[v];
        }
    }
    __syncthreads();

    // ---- Smoothed softmax over K per row (one thread per row) ----
    if (tid < ROWS) {
        const int r = tid;
        const float xnorm = xn[r*4] + xn[r*4+1] + xn[r*4+2] + xn[r*4+3];
        float mx = -3.4e38f;
        #pragma unroll
        for (int k = 0; k < K_; ++k) {
            float l = (xnorm - 2.0f * dotb[r * K_ + k] + cwn[k]) * sfl[k];
            wb[r * K_ + k] = l;
            mx = fmaxf(mx, l);
        }
        float s = 0.0f;
        #pragma unroll
        for (int k = 0; k < K_; ++k) {
            float e = __expf(wb[r * K_ + k] - mx);
            wb[r * K_ + k] = e;
            s += e;
        }
        const float inv = 1.0f / s;
        #pragma unroll
        for (int k = 0; k < K_; ++k) wb[r * K_ + k] *= inv;
    }
    __syncthreads();

    // ---- Column sums of W -> global wsum (needed for the -sum(w)*mu term) ----
    if (tid < K_) {
        float s = 0.0f;
        for (int r = 0; r < ROWS; ++r) s += wb[r * K_ + tid];
        atomicAdd(&wsum[b * K_ + tid], s);
    }

    // ---- GEMM2: encPart[K x C] = W^T[K x ROWS] * Xt[ROWS x C] ----
    // 2 M-tiles x 16 N-tiles = 32 tiles, 4 per wave.
    {
        const int kHi = (lane >> 4) * 2;
        for (int t = wave * 4; t < wave * 4 + 4; ++t) {
            const int mT = t >> 4;             // 0..1  (codeword tile)
            const int nT = t & 15;             // 0..15 (channel tile)
            const int colA = mT * 16 + (lane & 15);  // codeword index
            const int colB = nT * 16 + (lane & 15);  // channel index
            v8f acc = {};
            for (int kk = 0; kk < ROWS; kk += 4) {
                const int r0 = kk + kHi;
                v2f a, bf;
                a.x  = wb[ r0      * K_ + colA];
                a.y  = wb[(r0 + 1) * K_ + colA];
                bf.x = Xt[ r0      * C_ + colB];
                bf.y = Xt[(r0 + 1) * C_ + colB];
                acc = __builtin_amdgcn_wmma_f32_16x16x4_f32(
                    false, a, false, bf, (short)0, acc, false, false);
            }
            #pragma unroll
            for (int v = 0; v < 8; ++v) {
                int kc = mT * 16 + v + ((lane >> 4) << 3);
                int c  = nT * 16 + (lane & 15);
                atomicAdd(&enc[((long)b * K_ + kc) * C_ + c], acc[v]);
            }
        }
    }
}

// ---------------------------------------------------------------------------
// k2: enc <- enc - wsum*mu (the residual correction), accumulate BN stats.
// 128 blocks x 256 threads; each block owns 4 (b,k) slices, thread = channel.
// ---------------------------------------------------------------------------
__global__ __launch_bounds__(256)
void k2_correct(const float* __restrict__ cwg,
                float* __restrict__ enc,
                const float* __restrict__ wsum,
                float* __restrict__ sumc,
                float* __restrict__ sqc)
{
    const int c  = threadIdx.x;
    const int s0 = blockIdx.x * 4;
    float as = 0.0f, aq = 0.0f;
    #pragma unroll
    for (int i = 0; i < 4; ++i) {
        const int s = s0 + i;                  // slice = b*K + k
        const int k = s & (K_ - 1);
        float v = enc[(long)s * C_ + c] - wsum[s] * cwg[k * C_ + c];
        enc[(long)s * C_ + c] = v;
        as += v;
        aq += v * v;
    }
    atomicAdd(&sumc[c], as);
    atomicAdd(&sqc[c], aq);
}

// ---------------------------------------------------------------------------
// k3: BatchNorm (batch stats over (b,k)), ReLU, sum over codewords -> out.
// ---------------------------------------------------------------------------
__global__ __launch_bounds__(256)
void k3_out(const float* __restrict__ enc,
            const float* __restrict__ sumc,
            const float* __restrict__ sqc,
            const float* __restrict__ gamma,
            const float* __restrict__ beta,
            float* __restrict__ out)
{
    const int c = threadIdx.x;
    const int b = blockIdx.x;
    const float invN = 1.0f / (float)(B_ * K_);
    const float mean = sumc[c] * invN;
    const float var  = sqc[c] * invN - mean * mean;
    const float scl  = rsqrtf(var + BN_EPS) * gamma[c];
    const float bet  = beta[c];
    float acc = 0.0f;
    #pragma unroll
    for (int k = 0; k < K_; ++k) {
        float v = (enc[((long)b * K_ + k) * C_ + c] - mean) * scl + bet;
        acc += fmaxf(v, 0.0f);
    }
    out[b * C_ + c] = acc;
}

// ---------------------------------------------------------------------------
extern "C" void kernel_launch(void* const* d_in, const int* in_sizes, int n_in,
                              void* d_out, int out_size, void* d_ws, size_t ws_size,
                              hipStream_t stream) {
    const float* x     = (const float*)d_in[0];   // [16,32,32,256]
    const float* cw    = (const float*)d_in[1];   // [1,1,32,256]
    const float* sf    = (const float*)d_in[2];   // [1,1,32]
    const float* gamma = (const float*)d_in[3];   // [256]
    const float* beta  = (const float*)d_in[4];   // [256]
    float* out = (float*)d_out;                   // [16,256]
    float* ws  = (float*)d_ws;

    float* enc  = ws + WS_ENC;
    float* wsum = ws + WS_WSUM;
    float* sumc = ws + WS_SUM;
    float* sqc  = ws + WS_SQ;

    // zero accumulators
    k0_zero<<<(WS_TOTAL + 255) / 256, 256, 0, stream>>>(ws, WS_TOTAL);

    // fused assignment + aggregation (needs ~113 KB dynamic LDS; CDNA5 WGP
    // has 320 KB, so 2 workgroups/WGP)
    const size_t shmem = (size_t)(ROWS * C_ + K_ * C_ + 2 * ROWS * K_ +
                                  ROWS * 4 + 2 * K_) * sizeof(float);
    (void)hipFuncSetAttribute((const void*)k1_assign,
                              hipFuncAttributeMaxDynamicSharedMemorySize,
                              (int)shmem);
    k1_assign<<<NBLK1, 256, shmem, stream>>>(x, cw, sf, enc, wsum);

    // residual correction + BN statistics
    k2_correct<<<(B_ * K_) / 4, 256, 0, stream>>>(cw, enc, wsum, sumc, sqc);

    // BN + ReLU + reduce over K
    k3_out<<<B_, 256, 0, stream>>>(enc, sumc, sqc, gamma, beta, out);
}